// GCNNet_66726611911374
// MI455X (gfx1250) — compile-verified
//
#include <hip/hip_runtime.h>
#include <hip/hip_bf16.h>

typedef __attribute__((ext_vector_type(2))) float v2f;
typedef __attribute__((ext_vector_type(8))) float v8f;

// ---------------------------------------------------------------------------
// Degree accumulation: deg[r] += 1 for every edge source (self loop added later)
// ---------------------------------------------------------------------------
__global__ void deg_kernel(const long long* __restrict__ rows,
                           float* __restrict__ deg, long long E) {
    long long tid = (long long)blockIdx.x * blockDim.x + threadIdx.x;
    if (tid < E) {
        unsafeAtomicAdd(&deg[rows[tid]], 1.0f);
    }
}

// dinv[i] = rsqrt(deg[i] + 1)   (self loop contributes +1; always > 0)
__global__ void dinv_kernel(float* __restrict__ deg, int n) {
    int i = blockIdx.x * blockDim.x + threadIdx.x;
    if (i < n) {
        deg[i] = rsqrtf(deg[i] + 1.0f);
    }
}

// ---------------------------------------------------------------------------
// GEMM1: h[N,64] = x[N,256] @ W1[64,256]^T  via V_WMMA_F32_16X16X4_F32
// Block: 128 threads = 4 waves. Block owns 16 rows; wave w owns cols [16w,16w+16).
// A-frag (16x4 f32): lanes 0-15 -> row l, K = k+0/k+1 ; lanes 16-31 -> K = k+2/k+3.
// B-frag (4x16): lane l holds col n0+l; same K split by half -> contiguous float2
// from W1 row (since B[k][n] = W1[n][k]).
// ---------------------------------------------------------------------------
__global__ void gemm1_wmma(const float* __restrict__ x,
                           const float* __restrict__ w1,
                           float* __restrict__ h) {
    const int lane = threadIdx.x & 31;
    const int wave = threadIdx.x >> 5;      // 0..3 -> column tile
    const int half = lane >> 4;             // 0 or 1
    const int l    = lane & 15;
    const int row0 = blockIdx.x * 16;       // N % 16 == 0
    const int n0   = wave * 16;

    const float* arow = x  + (size_t)(row0 + l) * 256 + 2 * half;
    const float* brow = w1 + (size_t)(n0   + l) * 256 + 2 * half;

    v8f c = {};
#pragma unroll 8
    for (int k = 0; k < 256; k += 4) {
        v2f a = *(const v2f*)(arow + k);
        v2f b = *(const v2f*)(brow + k);
        c = __builtin_amdgcn_wmma_f32_16x16x4_f32(false, a, false, b,
                                                  (short)0, c, false, false);
    }

    // C/D layout: VGPR j -> lanes 0-15: M=j ; lanes 16-31: M=j+8 ; N = lane&15
    float* out = h + (size_t)(row0 + 8 * half) * 64 + n0 + l;
#pragma unroll
    for (int j = 0; j < 8; ++j) {
        out[(size_t)j * 64] = c[j];
    }
}

// ---------------------------------------------------------------------------
// SpMM over edges, feature dim 64: out[r,f] += dinv[r]*dinv[c] * feat[c,f]
// One thread per (edge, feature).
// ---------------------------------------------------------------------------
__global__ void spmm_edges64(const long long* __restrict__ rows,
                             const long long* __restrict__ cols,
                             const float* __restrict__ dinv,
                             const float* __restrict__ feat,
                             float* __restrict__ out) {
    long long tid = (long long)blockIdx.x * blockDim.x + threadIdx.x;
    long long e = tid >> 6;
    int f = (int)(tid & 63);
    long long r = rows[e];
    long long c = cols[e];
    float w = dinv[r] * dinv[c];
    unsafeAtomicAdd(&out[r * 64 + f], w * feat[c * 64 + f]);
}

// Self loop + ReLU (feature dim 64): hagg[i,f] = relu(hagg[i,f] + dinv[i]^2*h[i,f])
__global__ void selfloop_relu64(const float* __restrict__ dinv,
                                const float* __restrict__ h,
                                float* __restrict__ hagg, long long total) {
    long long idx = (long long)blockIdx.x * blockDim.x + threadIdx.x;
    if (idx < total) {
        long long i = idx >> 6;
        float d = dinv[i];
        float v = hagg[idx] + d * d * h[idx];
        hagg[idx] = v > 0.0f ? v : 0.0f;
    }
}

// ---------------------------------------------------------------------------
// GEMM2: h2[N,40] = hrelu[N,64] @ W2[40,64]^T. 3 col tiles (48 >= 40), B
// zero-padded for n >= 40, stores guarded. Block: 96 threads = 3 waves.
// ---------------------------------------------------------------------------
__global__ void gemm2_wmma(const float* __restrict__ hin,
                           const float* __restrict__ w2,
                           float* __restrict__ h2) {
    const int lane = threadIdx.x & 31;
    const int wave = threadIdx.x >> 5;      // 0..2 -> column tile
    const int half = lane >> 4;
    const int l    = lane & 15;
    const int row0 = blockIdx.x * 16;
    const int n0   = wave * 16;
    const int n    = n0 + l;                // B column owned by this lane

    const float* arow = hin + (size_t)(row0 + l) * 64 + 2 * half;
    const float* brow = w2  + (size_t)n * 64 + 2 * half;
    const bool bvalid = (n < 40);

    v8f c = {};
#pragma unroll
    for (int k = 0; k < 64; k += 4) {
        v2f a = *(const v2f*)(arow + k);
        v2f b;
        if (bvalid) {
            b = *(const v2f*)(brow + k);
        } else {
            b.x = 0.0f; b.y = 0.0f;
        }
        c = __builtin_amdgcn_wmma_f32_16x16x4_f32(false, a, false, b,
                                                  (short)0, c, false, false);
    }

    if (n < 40) {
        float* out = h2 + (size_t)(row0 + 8 * half) * 40 + n;
#pragma unroll
        for (int j = 0; j < 8; ++j) {
            out[(size_t)j * 40] = c[j];
        }
    }
}

// SpMM over edges, feature dim 40. blockDim = 320 -> 8 edges per block.
__global__ void spmm_edges40(const long long* __restrict__ rows,
                             const long long* __restrict__ cols,
                             const float* __restrict__ dinv,
                             const float* __restrict__ feat,
                             float* __restrict__ out) {
    long long e = (long long)blockIdx.x * 8 + (threadIdx.x / 40);
    int f = threadIdx.x % 40;
    long long r = rows[e];
    long long c = cols[e];
    float w = dinv[r] * dinv[c];
    unsafeAtomicAdd(&out[r * 40 + f], w * feat[c * 40 + f]);
}

// Self loop (feature dim 40), runs after all edge atomics have completed.
__global__ void selfloop40(const float* __restrict__ dinv,
                           const float* __restrict__ h2,
                           float* __restrict__ out, long long total) {
    long long idx = (long long)blockIdx.x * blockDim.x + threadIdx.x;
    if (idx < total) {
        long long i = idx / 40;
        float d = dinv[i];
        out[idx] += d * d * h2[idx];
    }
}

// ---------------------------------------------------------------------------
extern "C" void kernel_launch(void* const* d_in, const int* in_sizes, int n_in,
                              void* d_out, int out_size, void* d_ws, size_t ws_size,
                              hipStream_t stream) {
    const float*     x  = (const float*)d_in[0];        // [N,256]
    const float*     W1 = (const float*)d_in[1];        // [64,256]
    const float*     W2 = (const float*)d_in[2];        // [40,64]
    const long long* ei = (const long long*)d_in[3];    // [2,E] int64

    const long long N = in_sizes[0] / 256;              // 100000
    const long long E = in_sizes[3] / 2;                // 1600000

    const long long* rows = ei;
    const long long* cols = ei + E;

    // Workspace layout (floats): dinv[N] | h[N*64] | hagg[N*64] | h2[N*40]
    float* dinv = (float*)d_ws;
    float* h    = dinv + N;
    float* hagg = h + N * 64;
    float* h2   = hagg + N * 64;
    float* out  = (float*)d_out;

    // Zero accumulators (graph-capturable memset nodes)
    hipMemsetAsync(dinv, 0, (size_t)N * sizeof(float), stream);
    hipMemsetAsync(hagg, 0, (size_t)N * 64 * sizeof(float), stream);
    hipMemsetAsync(out,  0, (size_t)N * 40 * sizeof(float), stream);

    // 1. degrees -> dinv
    deg_kernel<<<(int)((E + 255) / 256), 256, 0, stream>>>(rows, dinv, E);
    dinv_kernel<<<(int)((N + 255) / 256), 256, 0, stream>>>(dinv, (int)N);

    // 2. h = x @ W1^T   (WMMA f32, 16-row tiles x 4 col-tiles per block)
    gemm1_wmma<<<(int)(N / 16), 128, 0, stream>>>(x, W1, h);

    // 3. hagg = A_norm @ h ; then self-loop + ReLU in place
    spmm_edges64<<<(int)(E * 64 / 256), 256, 0, stream>>>(rows, cols, dinv, h, hagg);
    selfloop_relu64<<<(int)(N * 64 / 256), 256, 0, stream>>>(dinv, h, hagg, N * 64);

    // 4. h2 = hagg @ W2^T (WMMA f32, 3 col-tiles, cols >= 40 masked)
    gemm2_wmma<<<(int)(N / 16), 96, 0, stream>>>(hagg, W2, h2);

    // 5. out = A_norm @ h2
    spmm_edges40<<<(int)(E / 8), 320, 0, stream>>>(rows, cols, dinv, h2, out);
    selfloop40<<<(int)((N * 40 + 255) / 256), 256, 0, stream>>>(dinv, h2, out, N * 40);
}